// GateNet_46755013984327
// MI455X (gfx1250) — compile-verified
//
#include <hip/hip_runtime.h>
#include <hip/hip_bf16.h>
#include <math.h>

// -------------------------------------------------------------------------
// GateNet pipeline for MI455X (gfx1250, wave32).
// Conv blocks: implicit-GEMM via V_WMMA_F32_16X16X4_F32 (fp32 matrix pipe).
//   M = flat output position (== NDHWC row index), N = Cout, K = 27*Cin.
//   Weight layout DHWIO == K-major x N, so B fragments are coalesced rows.
//   SAME padding for every block is pad_lo=0 / pad_hi=1 -> only upper check.
// v2: all geometry is compile-time (template), block-0 tap decode folds to
//     constants + cndmask, loads are clamped-address + cndmask (no exec
//     branches in the inner loop), NT=4 N-tiles for blocks 1-3.
// Workspace layout (floats):
//   h0 : [0, 28311552)          h1 : [28311552, 35389440)
//   h2 : [0, 1769472)  (h0 region reused)   h3 : [2097152, 2539520)
//   pooled : [4194304, 4196352)
// Peak requirement: ~142 MB of d_ws.
// -------------------------------------------------------------------------

typedef __attribute__((ext_vector_type(2))) float v2f;
typedef __attribute__((ext_vector_type(8))) float v8f;

#define BN_EPS 1e-3f

__device__ __forceinline__ float relu_(float v) { return v > 0.f ? v : 0.f; }

template <int CIN_LOG2, int SIN, int SOUT, int COUT, int NT>
__global__ __launch_bounds__(32) void conv3d_bn_relu_wmma(
    const float* __restrict__ in,    // [B, SIN,SIN,SIN, CIN] (NDHWC)
    const float* __restrict__ wgt,   // [27*CIN, COUT]        (DHWIO flat)
    const float* __restrict__ bias,
    const float* __restrict__ gamma, const float* __restrict__ beta,
    const float* __restrict__ mean,  const float* __restrict__ var,
    float* __restrict__ out)         // [M, COUT], M = 8*SOUT^3
{
    constexpr int CIN = 1 << CIN_LOG2;
    const int lane = threadIdx.x;          // wave32
    const int nlo  = lane & 15;
    const bool hi  = lane >= 16;
    const int koff = hi ? 2 : 0;           // K sub-offset within 4-wide chunk

    // A-fragment row = output position for this lane (compile-time SOUT ->
    // magic-multiply division)
    const int mrow = blockIdx.x * 16 + nlo;
    int t = mrow;
    const int ow = t % SOUT; t /= SOUT;
    const int oh = t % SOUT; t /= SOUT;
    const int od = t % SOUT; const int b = t / SOUT;
    const int id0 = od * 2, ih0 = oh * 2, iw0 = ow * 2;   // pad_lo == 0

    // NT N-columns (16-wide tiles each) per wave
    int ncol[NT];
    #pragma unroll
    for (int nt = 0; nt < NT; ++nt) ncol[nt] = blockIdx.y * (16 * NT) + nt * 16 + nlo;

    v8f acc[NT];
    #pragma unroll
    for (int nt = 0; nt < NT; ++nt) acc[nt] = (v8f){};

    if constexpr (CIN_LOG2 == 0) {
        // Block 0: K = 27, zero-padded to 28 (7 WMMA K-steps).
        // Tap coordinates are compile-time per (step, lane-half): one
        // cndmask each, no div/mod in the loop.
        const float* base = in + ((size_t)((b * SIN + id0) * SIN + ih0) * SIN + iw0);
        #pragma unroll
        for (int k0i = 0; k0i < 28; k0i += 4) {
            v2f a;
            v2f bf[NT];
            #pragma unroll
            for (int u = 0; u < 2; ++u) {
                const int kA = k0i + u;       // lanes 0-15
                const int kB = k0i + 2 + u;   // lanes 16-31
                const int kdA = kA / 9, khA = (kA / 3) % 3, kwA = kA % 3;
                const int kdB = kB / 9, khB = (kB / 3) % 3, kwB = kB % 3;
                const int offA = (kdA * SIN + khA) * SIN + kwA;
                const int offB = (kdB * SIN + khB) * SIN + kwB;
                const int kd = hi ? kdB : kdA;
                const int kh = hi ? khB : khA;
                const int kw = hi ? kwB : kwA;
                const int kk = hi ? kB : kA;
                const bool ok = (kk < 27) & (id0 + kd < SIN) & (ih0 + kh < SIN) &
                                (iw0 + kw < SIN);
                const int off = ok ? (hi ? offB : offA) : 0;  // clamped-safe addr
                const float lv = base[off];                    // unconditional load
                a[u] = ok ? lv : 0.f;
                #pragma unroll
                for (int nt = 0; nt < NT; ++nt) {
                    const float wv = wgt[(kk < 27 ? kk : 0) * COUT + ncol[nt]];
                    bf[nt][u] = (kk < 27) ? wv : 0.f;
                }
            }
            #pragma unroll
            for (int nt = 0; nt < NT; ++nt)
                acc[nt] = __builtin_amdgcn_wmma_f32_16x16x4_f32(
                    false, a, false, bf[nt], (short)0, acc[nt], false, false);
        }
    } else {
        // Cin >= 32: every 4-wide K chunk lives inside one spatial tap, so
        // hoist tap decode + bounds check out of the channel loop.
        for (int tap = 0; tap < 27; ++tap) {
            const int kw = tap % 3, kh = (tap / 3) % 3, kd = tap / 9;
            const int id = id0 + kd, ih = ih0 + kh, iw = iw0 + kw;
            const bool valid = (id < SIN) & (ih < SIN) & (iw < SIN);
            const size_t offRaw =
                ((size_t)((b * SIN + id) * SIN + ih) * SIN + iw) << CIN_LOG2;
            const float* abase = in + (valid ? offRaw : 0);   // clamped-safe addr
            const float* wbase = wgt + (size_t)(tap << CIN_LOG2) * COUT;
            #pragma unroll 8
            for (int ci0 = 0; ci0 < CIN; ci0 += 4) {
                const int kk = ci0 + koff;
                const float l0 = abase[kk];        // unconditional b64 pair
                const float l1 = abase[kk + 1];
                v2f a;
                a.x = valid ? l0 : 0.f;
                a.y = valid ? l1 : 0.f;
                const float* w0 = wbase + (size_t)kk * COUT;
                v2f bf[NT];
                #pragma unroll
                for (int nt = 0; nt < NT; ++nt) {
                    bf[nt].x = w0[ncol[nt]];
                    bf[nt].y = w0[COUT + ncol[nt]];
                }
                #pragma unroll
                for (int nt = 0; nt < NT; ++nt)
                    acc[nt] = __builtin_amdgcn_wmma_f32_16x16x4_f32(
                        false, a, false, bf[nt], (short)0, acc[nt], false, false);
            }
        }
    }

    // Fused bias + BN (inference) + ReLU epilogue
    const int mbase = blockIdx.x * 16 + (hi ? 8 : 0);  // M = v + 8*(lane>=16)
    #pragma unroll
    for (int nt = 0; nt < NT; ++nt) {
        const int n = ncol[nt];
        const float s = gamma[n] * rsqrtf(var[n] + BN_EPS);
        const float tt = (bias[n] - mean[n]) * s + beta[n];
        #pragma unroll
        for (int v = 0; v < 8; ++v) {
            out[(size_t)(mbase + v) * COUT + n] = relu_(acc[nt][v] * s + tt);
        }
    }
}

// Global average pool: h3 [8, P, 256] -> pooled [8, 256]
__global__ __launch_bounds__(256) void gap_kernel(const float* __restrict__ h,
                                                  float* __restrict__ pooled, int P)
{
    const int idx = blockIdx.x * blockDim.x + threadIdx.x;  // 0..2047
    const int c = idx & 255, bb = idx >> 8;
    const float* p = h + (size_t)bb * P * 256 + c;
    float s = 0.f;
    for (int i = 0; i < P; ++i) s += p[(size_t)i * 256];
    pooled[idx] = s / (float)P;
}

// Gating head: dense 256->256 + BN + ReLU, dense 256->8, top-2, softmax,
// transpose to [E, B]. One workgroup of 256 threads.
__global__ __launch_bounds__(256) void gate_kernel(
    const float* __restrict__ pooled,
    const float* __restrict__ w1, const float* __restrict__ b1,
    const float* __restrict__ g,  const float* __restrict__ be,
    const float* __restrict__ mu, const float* __restrict__ va,
    const float* __restrict__ w2, const float* __restrict__ b2,
    float* __restrict__ outp)
{
    __shared__ float sp[8 * 256];
    __shared__ float act[8 * 256];
    __shared__ float logits[64];
    const int c = threadIdx.x;

    for (int b = 0; b < 8; ++b) sp[b * 256 + c] = pooled[b * 256 + c];
    __syncthreads();

    float acc[8] = {0, 0, 0, 0, 0, 0, 0, 0};
    for (int k = 0; k < 256; ++k) {
        const float wv = w1[k * 256 + c];
        #pragma unroll
        for (int b = 0; b < 8; ++b) acc[b] += sp[b * 256 + k] * wv;
    }
    const float s = g[c] * rsqrtf(va[c] + BN_EPS);
    const float t = (b1[c] - mu[c]) * s + be[c];
    for (int b = 0; b < 8; ++b) act[b * 256 + c] = relu_(acc[b] * s + t);
    __syncthreads();

    if (c < 64) {
        const int e = c & 7, b = c >> 3;
        float a = b2[e];
        for (int k = 0; k < 256; ++k) a += act[b * 256 + k] * w2[k * 8 + e];
        logits[b * 8 + e] = a;
    }
    __syncthreads();

    if (c < 8) {
        const int b = c;
        float v[8], m1 = -INFINITY, m2 = -INFINITY;
        #pragma unroll
        for (int e = 0; e < 8; ++e) {
            v[e] = logits[b * 8 + e];
            if (v[e] > m1) { m2 = m1; m1 = v[e]; }
            else if (v[e] > m2) m2 = v[e];
        }
        // keep_top_k(k=2): mask = v >= 2nd-largest (ties kept, as in reference)
        float sum = 0.f, ex[8];
        #pragma unroll
        for (int e = 0; e < 8; ++e) {
            ex[e] = (v[e] >= m2) ? __expf(v[e] - m1) : 0.f;
            sum += ex[e];
        }
        #pragma unroll
        for (int e = 0; e < 8; ++e) outp[e * 8 + b] = ex[e] / sum;  // [E,B]
    }
}

extern "C" void kernel_launch(void* const* d_in, const int* in_sizes, int n_in,
                              void* d_out, int out_size, void* d_ws, size_t ws_size,
                              hipStream_t stream)
{
    (void)in_sizes; (void)n_in; (void)out_size; (void)ws_size;

    // setup_inputs() dict order, lists flattened in sequence:
    const float* x = (const float*)d_in[0];
    const float* bk[4], * bb[4], * bg[4], * bbe[4], * bm[4], * bv[4];
    for (int i = 0; i < 4; ++i) {
        bk[i]  = (const float*)d_in[1 + i];
        bb[i]  = (const float*)d_in[5 + i];
        bg[i]  = (const float*)d_in[9 + i];
        bbe[i] = (const float*)d_in[13 + i];
        bm[i]  = (const float*)d_in[17 + i];
        bv[i]  = (const float*)d_in[21 + i];
    }
    const float* w1k = (const float*)d_in[25];
    const float* w1b = (const float*)d_in[26];
    const float* wg  = (const float*)d_in[27];
    const float* wbe = (const float*)d_in[28];
    const float* wmu = (const float*)d_in[29];
    const float* wva = (const float*)d_in[30];
    const float* w2k = (const float*)d_in[31];
    const float* w2b = (const float*)d_in[32];

    float* ws = (float*)d_ws;
    float* h0 = ws;                    // 28,311,552 f
    float* h1 = ws + 28311552u;        //  7,077,888 f
    float* h2 = ws;                    //  1,769,472 f (reuse h0 region)
    float* h3 = ws + 2097152u;         //    442,368 f
    float* pooled = ws + 4194304u;     //      2,048 f

    const dim3 wave(32, 1, 1);
    // grid.x = M/16 tiles, grid.y = COUT/(16*NT)
    conv3d_bn_relu_wmma<0, 96, 48, 32, 2><<<dim3(55296, 1, 1), wave, 0, stream>>>(
        x,  bk[0], bb[0], bg[0], bbe[0], bm[0], bv[0], h0);
    conv3d_bn_relu_wmma<5, 48, 24, 64, 4><<<dim3(6912, 1, 1), wave, 0, stream>>>(
        h0, bk[1], bb[1], bg[1], bbe[1], bm[1], bv[1], h1);
    conv3d_bn_relu_wmma<6, 24, 12, 128, 4><<<dim3(864, 2, 1), wave, 0, stream>>>(
        h1, bk[2], bb[2], bg[2], bbe[2], bm[2], bv[2], h2);
    conv3d_bn_relu_wmma<7, 12, 6, 256, 4><<<dim3(108, 4, 1), wave, 0, stream>>>(
        h2, bk[3], bb[3], bg[3], bbe[3], bm[3], bv[3], h3);

    gap_kernel<<<dim3(8, 1, 1), dim3(256, 1, 1), 0, stream>>>(h3, pooled, 216);
    gate_kernel<<<dim3(1, 1, 1), dim3(256, 1, 1), 0, stream>>>(
        pooled, w1k, w1b, wg, wbe, wmu, wva, w2k, w2b, (float*)d_out);
}